// ReflexNet_8246337208562
// MI455X (gfx1250) — compile-verified
//
#include <hip/hip_runtime.h>

typedef float v2f __attribute__((ext_vector_type(2)));
typedef float v4f __attribute__((ext_vector_type(4)));
typedef float v8f __attribute__((ext_vector_type(8)));

// D = A(16x4 f32) * B(4x16 f32) + C(16x16 f32), wave32, exact fp32.
#define WMMA4(a, b, c) \
  __builtin_amdgcn_wmma_f32_16x16x4_f32(false, (a), false, (b), (short)0, (c), false, false)

#if __has_builtin(__builtin_amdgcn_tanhf)
__device__ __forceinline__ float tanh_fast(float x) { return __builtin_amdgcn_tanhf(x); }
#else
__device__ __forceinline__ float tanh_fast(float x) { return tanhf(x); }
#endif
// sigmoid(x) = 0.5*tanh(x/2) + 0.5
__device__ __forceinline__ float sigf(float x) { return fmaf(tanh_fast(0.5f * x), 0.5f, 0.5f); }

// Row permutation: physical WMMA row q (0..15) holds logical row
// pi(q) = sigma(q&7) + 2*(q>>3), sigma(v) = 4*(v>>1) + (v&1).
// With this, a D-tile's even-aligned register pairs (d[2j],d[2j+1]) ARE the
// K=4 B-operand slices of the next matmul: no cross-lane data movement.
__device__ __forceinline__ int pi_row(int q) {
  return 4 * ((q & 7) >> 1) + (q & 1) + 2 * (q >> 3);
}

__device__ __forceinline__ v8f loadv8(const float* p) {
  v4f a = *(const v4f*)p;
  v4f b = *(const v4f*)(p + 4);
  v8f r;
  r[0] = a[0]; r[1] = a[1]; r[2] = a[2]; r[3] = a[3];
  r[4] = b[0]; r[5] = b[1]; r[6] = b[2]; r[7] = b[3];
  return r;
}

__device__ __forceinline__ v2f pairof(const v8f& d, int j) {
  v2f r; r[0] = d[2 * j]; r[1] = d[2 * j + 1]; return r;
}

__global__ __launch_bounds__(32, 1) void lstm_fused_wmma(
    const float* __restrict__ x,   // [B,T,12]
    const float* __restrict__ h0,  // [1,B,16]
    const float* __restrict__ c0,  // [1,B,16]
    const float* __restrict__ W1,  // [32,12]
    const float* __restrict__ b1,  // [32]
    const float* __restrict__ Wih, // [64,32]
    const float* __restrict__ Whh, // [64,16]
    const float* __restrict__ bih, // [64]
    const float* __restrict__ bhh, // [64]
    const float* __restrict__ W2,  // [1,16]
    const float* __restrict__ b2,  // [1]
    float* __restrict__ out)       // [4096] out | [4096*16] hT | [4096*16] cT
{
  constexpr int T = 512, IN = 12, BATCH = 4096, H = 16;
  __shared__ float ldsb[96];  // pre-permuted: [0..63] bih+bhh, [64..95] b1

  const int  L   = threadIdx.x;
  const int  n   = L & 15;          // WMMA row (A) / batch column (B,C,D)
  const bool lo  = (L < 16);
  const int  hi2 = lo ? 0 : 2;      // K offset for A/B high half
  const int  hi8 = lo ? 0 : 8;      // M offset for C/D high half
  const int  bb  = blockIdx.x * 16 + n;
  const int  pn  = pi_row(n);       // logical weight row for this lane

  // Pre-permuted bias tables (C-initializers read contiguously per step)
  for (int p = L; p < 64; p += 32) {
    int tt = p >> 4, q = p & 15;
    int l = 16 * tt + pi_row(q);
    ldsb[p] = bih[l] + bhh[l];
  }
  if (L < 32) {
    int mt = L >> 4, q = L & 15;
    ldsb[64 + L] = b1[16 * mt + pi_row(q)];
  }
  __syncthreads();

  // ---- A-operand weights, rows permuted by pi ----
  v2f aU[2][3];                           // W1: 2 M-tiles x 3 K-slices
#pragma unroll
  for (int mt = 0; mt < 2; ++mt)
#pragma unroll
    for (int j = 0; j < 3; ++j)
      aU[mt][j] = *(const v2f*)(W1 + (mt * 16 + pn) * 12 + 4 * j + hi2);

  v2f aI[4][8];                           // W_ih: 4 gate tiles x 8 K-slices
#pragma unroll
  for (int tt = 0; tt < 4; ++tt)
#pragma unroll
    for (int j = 0; j < 8; ++j)
      aI[tt][j] = *(const v2f*)(Wih + (tt * 16 + pn) * 32 + 4 * j + hi2);

  v2f aH[4][4];                           // W_hh: 4 gate tiles x 4 K-slices
#pragma unroll
  for (int tt = 0; tt < 4; ++tt)
#pragma unroll
    for (int j = 0; j < 4; ++j)
      aH[tt][j] = *(const v2f*)(Whh + (tt * 16 + pn) * 16 + 4 * j + hi2);

  // ---- State in permuted C/D layout: elem v <-> logical index sigma(v)+hi2 ----
  v8f ct, hD;
#pragma unroll
  for (int j = 0; j < 4; ++j) {           // sigma(2j)=4j, sigma(2j+1)=4j+1
    v2f tc = *(const v2f*)(c0 + bb * H + 4 * j + hi2);
    v2f th = *(const v2f*)(h0 + bb * H + 4 * j + hi2);
    ct[2 * j] = tc[0]; ct[2 * j + 1] = tc[1];
    hD[2 * j] = th[0]; hD[2 * j + 1] = th[1];
  }

  const float* xp = x + (size_t)bb * T * IN + hi2;
  v2f xB0 = *(const v2f*)(xp);
  v2f xB1 = *(const v2f*)(xp + 4);
  v2f xB2 = *(const v2f*)(xp + 8);

#pragma unroll 1
  for (int t = 0; t < T; ++t) {
    // Prefetch next timestep's x row (double-buffered; last iter reloads same row)
    const float* pnext = (t == T - 1) ? xp : (xp + IN);
    v2f nx0 = *(const v2f*)(pnext);
    v2f nx1 = *(const v2f*)(pnext + 4);
    v2f nx2 = *(const v2f*)(pnext + 8);
    __builtin_prefetch(pnext + 4 * IN, 0, 3);
    xp = pnext;

    // Bias C-initializers (pre-permuted, contiguous)
    v8f u0 = loadv8(&ldsb[64 + hi8]);
    v8f u1 = loadv8(&ldsb[64 + 16 + hi8]);
    v8f g0 = loadv8(&ldsb[0 + hi8]);
    v8f g1 = loadv8(&ldsb[16 + hi8]);
    v8f g2 = loadv8(&ldsb[32 + hi8]);
    v8f g3 = loadv8(&ldsb[48 + hi8]);

    // u^T = relu(W1p @ x^T + b1p): two 16x16 tiles (rows permuted by pi)
    u0 = WMMA4(aU[0][0], xB0, u0);
    u1 = WMMA4(aU[1][0], xB0, u1);
    u0 = WMMA4(aU[0][1], xB1, u0);
    u1 = WMMA4(aU[1][1], xB1, u1);
    u0 = WMMA4(aU[0][2], xB2, u0);
    u1 = WMMA4(aU[1][2], xB2, u1);
#pragma unroll
    for (int v = 0; v < 8; ++v) {
      u0[v] = fmaxf(u0[v], 0.0f);
      u1[v] = fmaxf(u1[v], 0.0f);
    }

    // gates^T: accumulator register pairs serve directly as B-slices
#pragma unroll
    for (int j = 0; j < 4; ++j) {
      v2f bs = pairof(u0, j);             // u logical rows 4j..4j+3
      g0 = WMMA4(aI[0][j], bs, g0);
      g1 = WMMA4(aI[1][j], bs, g1);
      g2 = WMMA4(aI[2][j], bs, g2);
      g3 = WMMA4(aI[3][j], bs, g3);
    }
#pragma unroll
    for (int j = 0; j < 4; ++j) {
      v2f bs = pairof(u1, j);             // u logical rows 16+4j..
      g0 = WMMA4(aI[0][4 + j], bs, g0);
      g1 = WMMA4(aI[1][4 + j], bs, g1);
      g2 = WMMA4(aI[2][4 + j], bs, g2);
      g3 = WMMA4(aI[3][4 + j], bs, g3);
    }
#pragma unroll
    for (int j = 0; j < 4; ++j) {
      v2f bs = pairof(hD, j);             // h logical rows 4j..4j+3
      g0 = WMMA4(aH[0][j], bs, g0);
      g1 = WMMA4(aH[1][j], bs, g1);
      g2 = WMMA4(aH[2][j], bs, g2);
      g3 = WMMA4(aH[3][j], bs, g3);
    }

    // LSTM cell, elementwise in the shared permuted layout (i,f,g,o)
#pragma unroll
    for (int v = 0; v < 8; ++v) {
      float it = sigf(g0[v]);
      float ft = sigf(g1[v]);
      float gt = tanh_fast(g2[v]);
      float ot = sigf(g3[v]);
      float c  = ft * ct[v] + it * gt;
      ct[v] = c;
      hD[v] = ot * tanh_fast(c);
    }

    xB0 = nx0; xB1 = nx1; xB2 = nx2;
  }

  // ---- head: out[b] = tanh(hT . W2 + b2) ----
  float part = 0.0f;
#pragma unroll
  for (int j = 0; j < 4; ++j) {
    v2f w2p = *(const v2f*)(W2 + 4 * j + hi2);   // W2[sigma(2j)+hi2 ..]
    part = fmaf(hD[2 * j], w2p[0], part);
    part = fmaf(hD[2 * j + 1], w2p[1], part);
  }
  float tot = part + __shfl_xor(part, 16, 32);
  if (lo) out[bb] = tanh_fast(tot + b2[0]);

  // hT / cT: un-permute; sigma keeps pairs contiguous -> 4 float2 stores each
  float* hp = out + BATCH + bb * H;
  float* cp = out + BATCH + BATCH * H + bb * H;
#pragma unroll
  for (int j = 0; j < 4; ++j) {
    v2f th, tc;
    th[0] = hD[2 * j]; th[1] = hD[2 * j + 1];
    tc[0] = ct[2 * j]; tc[1] = ct[2 * j + 1];
    *(v2f*)(hp + 4 * j + hi2) = th;
    *(v2f*)(cp + 4 * j + hi2) = tc;
  }
}

extern "C" void kernel_launch(void* const* d_in, const int* in_sizes, int n_in,
                              void* d_out, int out_size, void* d_ws, size_t ws_size,
                              hipStream_t stream) {
  const float* x   = (const float*)d_in[0];
  const float* h0  = (const float*)d_in[1];
  const float* c0  = (const float*)d_in[2];
  const float* W1  = (const float*)d_in[3];
  const float* b1  = (const float*)d_in[4];
  const float* Wih = (const float*)d_in[5];
  const float* Whh = (const float*)d_in[6];
  const float* bih = (const float*)d_in[7];
  const float* bhh = (const float*)d_in[8];
  const float* W2  = (const float*)d_in[9];
  const float* b2  = (const float*)d_in[10];
  float* out = (float*)d_out;

  dim3 grid(4096 / 16), block(32);
  lstm_fused_wmma<<<grid, block, 0, stream>>>(x, h0, c0, W1, b1, Wih, Whh,
                                              bih, bhh, W2, b2, out);
}